// GAT_72310069395897
// MI455X (gfx1250) — compile-verified
//
#include <hip/hip_runtime.h>
#include <math.h>

typedef __attribute__((ext_vector_type(2))) float v2f;
typedef __attribute__((ext_vector_type(8))) float v8f;

namespace {
constexpr int kN  = 50000;   // nodes
constexpr int kE  = 800000;  // edges (before self loops)
constexpr int kF  = 128;     // IN == H*HID == H*OUT
constexpr int kH  = 4;       // heads
constexpr int kC  = 32;      // channels per head
constexpr int kG  = 64;      // graphs
constexpr int kEL = kE + kN; // edges incl. self loops
constexpr float kNegSlope = 0.2f;
}

// ---------------------------------------------------------------------------
// float atomic max via signed/unsigned integer atomics (IEEE ordering trick).
// Lowers to GLOBAL_ATOMIC_MAX_I32 / GLOBAL_ATOMIC_MIN_U32 (both exist on CDNA5).
// ---------------------------------------------------------------------------
__device__ __forceinline__ void atomic_max_f32(float* addr, float val) {
  if (val >= 0.0f) {
    atomicMax((int*)addr, __float_as_int(val));
  } else {
    atomicMin((unsigned int*)addr, __float_as_uint(val));
  }
}

// ---------------------------------------------------------------------------
// GEMM: Out[n,128] = X[n,128] @ W[128,128] using V_WMMA_F32_16X16X4_F32.
// One wave computes a 16-row x 128-col tile (8 accumulators of v8f).
// A frag (16x4 f32): lane[0:15]=M rows, VGPR0=K(2*half), VGPR1=K(2*half+1).
// B frag (4x16 f32): VGPR j = row K=j (+2 for lanes 16-31), N = lane&15.
// C/D (16x16 f32): VGPR j = row M=j (+8 for lanes 16-31), N = lane&15.
// ---------------------------------------------------------------------------
__global__ __launch_bounds__(256) void gemm128_wmma_f32(
    const float* __restrict__ X, const float* __restrict__ W,
    float* __restrict__ Out, int nrows) {
  const int wave = blockIdx.x * (blockDim.x >> 5) + (threadIdx.x >> 5);
  const int ntiles = nrows >> 4;           // nrows divisible by 16
  if (wave >= ntiles) return;              // wave-uniform exit, EXEC stays full
  const int lane = threadIdx.x & 31;
  const int half = lane >> 4;              // 0: lanes 0-15, 1: lanes 16-31
  const int r    = lane & 15;

  const float* __restrict__ xrow = X + (size_t)(wave * 16 + r) * kF;

  v8f acc[8];
#pragma unroll
  for (int t = 0; t < 8; ++t)
#pragma unroll
    for (int j = 0; j < 8; ++j) acc[t][j] = 0.0f;

  for (int k = 0; k < kF; k += 4) {
    v2f a;
    a.x = xrow[k + 2 * half];
    a.y = xrow[k + 2 * half + 1];
#pragma unroll
    for (int t = 0; t < 8; ++t) {
      v2f b;
      b.x = W[(size_t)(k + 2 * half) * kF + t * 16 + r];
      b.y = W[(size_t)(k + 2 * half + 1) * kF + t * 16 + r];
      acc[t] = __builtin_amdgcn_wmma_f32_16x16x4_f32(
          /*neg_a=*/false, a, /*neg_b=*/false, b,
          /*c_mod=*/(short)0, acc[t], /*reuse_a=*/false, /*reuse_b=*/false);
    }
  }

#pragma unroll
  for (int t = 0; t < 8; ++t)
#pragma unroll
    for (int j = 0; j < 8; ++j) {
      const int orow = wave * 16 + j + 8 * half;
      Out[(size_t)orow * kF + t * 16 + r] = acc[t][j];
    }
}

// ---------------------------------------------------------------------------
// Per-node attention logit halves: as[n,h] = <h[n,h,:], att_src[h,:]>, same ad.
// ---------------------------------------------------------------------------
__global__ __launch_bounds__(256) void alpha_kernel(
    const float* __restrict__ hfeat, const float* __restrict__ att_s,
    const float* __restrict__ att_d, float* __restrict__ as_,
    float* __restrict__ ad_) {
  const int idx = blockIdx.x * blockDim.x + threadIdx.x;  // n*H
  if (idx >= kN * kH) return;
  const int node = idx >> 2, h = idx & 3;
  const float* __restrict__ hp = hfeat + (size_t)node * kF + h * kC;
  float s1 = 0.f, s2 = 0.f;
#pragma unroll
  for (int c = 0; c < kC; ++c) {
    const float v = hp[c];
    s1 += v * att_s[h * kC + c];
    s2 += v * att_d[h * kC + c];
  }
  as_[idx] = s1;
  ad_[idx] = s2;
}

// ---------------------------------------------------------------------------
// Edge helpers: thread/(wave) per (edge, head); self loops appended at e>=kE.
// ---------------------------------------------------------------------------
__device__ __forceinline__ void edge_ends(const int* __restrict__ ei,
                                          long long e, int& src, int& dst) {
  if (e < kE) { src = ei[e]; dst = ei[kE + e]; }
  else        { src = dst = (int)(e - kE); }
}

__global__ __launch_bounds__(256) void edge_max_kernel(
    const int* __restrict__ ei, const float* __restrict__ as_,
    const float* __restrict__ ad_, float* __restrict__ m_) {
  const long long idx = (long long)blockIdx.x * blockDim.x + threadIdx.x;
  if (idx >= (long long)kEL * kH) return;
  const int h = (int)(idx & 3);
  int src, dst;
  edge_ends(ei, idx >> 2, src, dst);
  float v = as_[src * kH + h] + ad_[dst * kH + h];
  v = v > 0.f ? v : kNegSlope * v;          // leaky_relu
  atomic_max_f32(&m_[dst * kH + h], v);
}

__global__ __launch_bounds__(256) void edge_sum_kernel(
    const int* __restrict__ ei, const float* __restrict__ as_,
    const float* __restrict__ ad_, const float* __restrict__ m_,
    float* __restrict__ s_) {
  const long long idx = (long long)blockIdx.x * blockDim.x + threadIdx.x;
  if (idx >= (long long)kEL * kH) return;
  const int h = (int)(idx & 3);
  int src, dst;
  edge_ends(ei, idx >> 2, src, dst);
  float v = as_[src * kH + h] + ad_[dst * kH + h];
  v = v > 0.f ? v : kNegSlope * v;
  atomicAdd(&s_[dst * kH + h], __expf(v - m_[dst * kH + h]));
}

// One wave per (edge, head); lane == channel -> coalesced gather + scatter.
__global__ __launch_bounds__(256) void edge_agg_kernel(
    const int* __restrict__ ei, const float* __restrict__ as_,
    const float* __restrict__ ad_, const float* __restrict__ m_,
    const float* __restrict__ s_, const float* __restrict__ hsrc,
    float* __restrict__ agg) {
  const long long wid =
      (long long)blockIdx.x * (blockDim.x >> 5) + (threadIdx.x >> 5);
  if (wid >= (long long)kEL * kH) return;
  const int lane = threadIdx.x & 31;
  const int h = (int)(wid & 3);
  int src, dst;
  edge_ends(ei, wid >> 2, src, dst);
  float v = as_[src * kH + h] + ad_[dst * kH + h];
  v = v > 0.f ? v : kNegSlope * v;
  const float coef =
      __expf(v - m_[dst * kH + h]) / (s_[dst * kH + h] + 1e-16f);
  const float val = coef * hsrc[(size_t)src * kF + h * kC + lane];
  atomicAdd(&agg[(size_t)dst * kF + h * kC + lane], val);
}

// ---------------------------------------------------------------------------
// bias add (+ optional ELU)
// ---------------------------------------------------------------------------
__global__ __launch_bounds__(256) void bias_act_kernel(
    const float* __restrict__ agg, const float* __restrict__ bias,
    float* __restrict__ out, int do_elu) {
  const long long idx = (long long)blockIdx.x * blockDim.x + threadIdx.x;
  if (idx >= (long long)kN * kF) return;
  float v = agg[idx] + bias[idx & (kF - 1)];
  if (do_elu) v = v > 0.f ? v : (__expf(v) - 1.0f);
  out[idx] = v;
}

__global__ __launch_bounds__(256) void fill_kernel(float* __restrict__ p,
                                                   float v, long long n) {
  const long long idx = (long long)blockIdx.x * blockDim.x + threadIdx.x;
  if (idx < n) p[idx] = v;
}

// ---------------------------------------------------------------------------
// Pooling: counts, mean-sum, max per graph (batch sorted, but atomics are fine
// since everything stays L2-resident).
// ---------------------------------------------------------------------------
__global__ __launch_bounds__(256) void count_kernel(
    const int* __restrict__ batch, float* __restrict__ cnt) {
  const int i = blockIdx.x * blockDim.x + threadIdx.x;
  if (i < kN) atomicAdd(&cnt[batch[i]], 1.0f);
}

__global__ __launch_bounds__(256) void pool_kernel(
    const float* __restrict__ h2, const int* __restrict__ batch,
    float* __restrict__ msum, float* __restrict__ mmax) {
  const long long idx = (long long)blockIdx.x * blockDim.x + threadIdx.x;
  if (idx >= (long long)kN * kF) return;
  const int node = (int)(idx >> 7), f = (int)(idx & (kF - 1));
  const int g = batch[node];
  const float v = h2[idx];
  atomicAdd(&msum[g * kF + f], v);
  atomic_max_f32(&mmax[g * kF + f], v);
}

__global__ __launch_bounds__(128) void final_kernel(
    const float* __restrict__ msum, const float* __restrict__ mmax,
    const float* __restrict__ cnt, const float* __restrict__ lin_w,
    const float* __restrict__ lin_b, float* __restrict__ out) {
  const int g = blockIdx.x, f = threadIdx.x;  // 128 threads
  const float denom = fmaxf(cnt[g], 1.0f);
  float mx = mmax[g * kF + f];
  if (!isfinite(mx)) mx = 0.0f;
  const float pooled = msum[g * kF + f] / denom + mx;
  __shared__ float red[kF];
  red[f] = pooled * lin_w[f];
  __syncthreads();
  for (int s = kF / 2; s > 0; s >>= 1) {
    if (f < s) red[f] += red[f + s];
    __syncthreads();
  }
  if (f == 0) out[g] = red[0] + lin_b[0];
}

// ---------------------------------------------------------------------------
// Orchestration
// ---------------------------------------------------------------------------
extern "C" void kernel_launch(void* const* d_in, const int* in_sizes, int n_in,
                              void* d_out, int out_size, void* d_ws,
                              size_t ws_size, hipStream_t stream) {
  (void)in_sizes; (void)n_in; (void)out_size; (void)ws_size;
  const float* x        = (const float*)d_in[0];
  const int*   ei       = (const int*)d_in[1];   // [2,E] row-major
  const int*   batch    = (const int*)d_in[2];
  const float* W1       = (const float*)d_in[3];
  const float* att_src1 = (const float*)d_in[4];
  const float* att_dst1 = (const float*)d_in[5];
  const float* b1       = (const float*)d_in[6];
  const float* W2       = (const float*)d_in[7];
  const float* att_src2 = (const float*)d_in[8];
  const float* att_dst2 = (const float*)d_in[9];
  const float* b2       = (const float*)d_in[10];
  const float* lin_w    = (const float*)d_in[11];
  const float* lin_b    = (const float*)d_in[12];
  float* out = (float*)d_out;

  float* ws   = (float*)d_ws;
  float* bufA = ws;                       // N*128
  float* bufB = bufA + (size_t)kN * kF;   // N*128
  float* as_  = bufB + (size_t)kN * kF;   // N*H
  float* ad_  = as_ + kN * kH;
  float* m_   = ad_ + kN * kH;
  float* s_   = m_ + kN * kH;
  float* msum = s_ + kN * kH;             // G*128
  float* mmax = msum + kG * kF;           // G*128
  float* cnt  = mmax + kG * kF;           // G

  const long long nf  = (long long)kN * kF;       // 6.4M
  const long long nh  = (long long)kN * kH;       // 200K
  const long long eh  = (long long)kEL * kH;      // 3.4M
  const int gemm_blk  = (int)(((kN / 16) * 32 + 255) / 256);
  const int nf_blk    = (int)((nf + 255) / 256);
  const int nh_blk    = (int)((nh + 255) / 256);
  const int eh_blk    = (int)((eh + 255) / 256);
  const int agg_blk   = (int)((eh + 7) / 8);      // 8 waves per 256-thr block

  // ---- layer 1: h = x @ W1 -> bufA -------------------------------------
  gemm128_wmma_f32<<<gemm_blk, 256, 0, stream>>>(x, W1, bufA, kN);
  alpha_kernel<<<nh_blk, 256, 0, stream>>>(bufA, att_src1, att_dst1, as_, ad_);
  fill_kernel<<<nh_blk, 256, 0, stream>>>(m_, -INFINITY, nh);
  fill_kernel<<<nh_blk, 256, 0, stream>>>(s_, 0.0f, nh);
  fill_kernel<<<nf_blk, 256, 0, stream>>>(bufB, 0.0f, nf);
  edge_max_kernel<<<eh_blk, 256, 0, stream>>>(ei, as_, ad_, m_);
  edge_sum_kernel<<<eh_blk, 256, 0, stream>>>(ei, as_, ad_, m_, s_);
  edge_agg_kernel<<<agg_blk, 256, 0, stream>>>(ei, as_, ad_, m_, s_, bufA, bufB);
  bias_act_kernel<<<nf_blk, 256, 0, stream>>>(bufB, b1, bufA, /*elu=*/1);

  // ---- layer 2: h2 = gat(bufA) -----------------------------------------
  gemm128_wmma_f32<<<gemm_blk, 256, 0, stream>>>(bufA, W2, bufB, kN);
  alpha_kernel<<<nh_blk, 256, 0, stream>>>(bufB, att_src2, att_dst2, as_, ad_);
  fill_kernel<<<nh_blk, 256, 0, stream>>>(m_, -INFINITY, nh);
  fill_kernel<<<nh_blk, 256, 0, stream>>>(s_, 0.0f, nh);
  fill_kernel<<<nf_blk, 256, 0, stream>>>(bufA, 0.0f, nf);
  edge_max_kernel<<<eh_blk, 256, 0, stream>>>(ei, as_, ad_, m_);
  edge_sum_kernel<<<eh_blk, 256, 0, stream>>>(ei, as_, ad_, m_, s_);
  edge_agg_kernel<<<agg_blk, 256, 0, stream>>>(ei, as_, ad_, m_, s_, bufB, bufA);
  bias_act_kernel<<<nf_blk, 256, 0, stream>>>(bufA, b2, bufA, /*elu=*/0);

  // ---- pooling + linear -------------------------------------------------
  const int gf_blk = (kG * kF + 255) / 256;
  fill_kernel<<<gf_blk, 256, 0, stream>>>(msum, 0.0f, kG * kF);
  fill_kernel<<<gf_blk, 256, 0, stream>>>(mmax, -INFINITY, kG * kF);
  fill_kernel<<<1, 256, 0, stream>>>(cnt, 0.0f, kG);
  count_kernel<<<(kN + 255) / 256, 256, 0, stream>>>(batch, cnt);
  pool_kernel<<<nf_blk, 256, 0, stream>>>(bufA, batch, msum, mmax);
  final_kernel<<<kG, kF, 0, stream>>>(msum, mmax, cnt, lin_w, lin_b, out);
}